// TTACont_27127013441911
// MI455X (gfx1250) — compile-verified
//
#include <hip/hip_runtime.h>
#include <cmath>
#include <cstdint>

// ---------------------------------------------------------------------------
// Problem constants (from the reference):  S is [2048, 32768] f32.
//   loss = -mean_row( r * log(r + 1e-10) ),  r = (sum of top-10 sigmoid) / rowsum
// Monotonicity: top-10 of normalized sigmoid == sigmoid of top-10 raw values.
// ---------------------------------------------------------------------------
namespace {
constexpr int   kB       = 2048;
constexpr int   kN       = 32768;
constexpr int   kThreads = 256;                    // 8 wave32 per block
constexpr int   kWaves   = kThreads / 32;
constexpr int   kChunks  = kN / (kThreads * 4);    // 32 float4 chunks / thread
constexpr int   kK       = 10;
constexpr float kExpC    = -0.57707801635558534f;  // -1 / (2.5 * ln2)
static_assert(kChunks == 32, "pipeline grouping assumes 32 chunks");
}

// CDNA5 async global->LDS path (ASYNCcnt-tracked). Guarded so the file also
// compiles on toolchains without the builtins (falls back to direct loads).
#if defined(__gfx1250__) && __has_builtin(__builtin_amdgcn_global_load_async_to_lds_b128)
  #define HAS_ASYNC_LDS 1
#else
  #define HAS_ASYNC_LDS 0
#endif

#if HAS_ASYNC_LDS
  #if __has_builtin(__builtin_amdgcn_s_wait_asynccnt)
    #define WAIT_ASYNC(n) do { __builtin_amdgcn_s_wait_asynccnt(n); \
                               asm volatile("" ::: "memory"); } while (0)
  #else
    #define WAIT_ASYNC(n) asm volatile("s_wait_asynccnt " #n ::: "memory")
  #endif
#endif

__device__ __forceinline__ float sig_fast(float x) {
#if defined(__HIP_DEVICE_COMPILE__)
  // sigmoid(x/2.5) = 1 / (1 + exp2(x * -1/(2.5*ln2)))  -> v_exp_f32 + v_rcp_f32
  float e = __builtin_amdgcn_exp2f(x * kExpC);
  return __builtin_amdgcn_rcpf(1.0f + e);
#else
  (void)x; return 0.0f;
#endif
}

__device__ __forceinline__ float wave_max(float v) {
  #pragma unroll
  for (int off = 16; off > 0; off >>= 1) v = fmaxf(v, __shfl_xor(v, off, 32));
  return v;
}
__device__ __forceinline__ float wave_sum(float v) {
  #pragma unroll
  for (int off = 16; off > 0; off >>= 1) v += __shfl_xor(v, off, 32);
  return v;
}

__global__ void __launch_bounds__(kThreads)
row_loss_kernel(const float* __restrict__ S, float* __restrict__ terms) {
  extern __shared__ float4 sdata[];            // full row: 128 KB (CDNA5 LDS)
  __shared__ float s_sum[kWaves];
  __shared__ float s_top[kWaves * kK];

  const int tid  = threadIdx.x;
  const int lane = tid & 31;
  const int wave = tid >> 5;
  const int row  = blockIdx.x;

  const float4* __restrict__ g = (const float4*)(S + (size_t)row * kN);

#if HAS_ASYNC_LDS
  // Issue the whole row (32 async b128 loads / thread, ASYNCcnt <= 32 < 63).
  {
    typedef int v4i __attribute__((vector_size(16)));
    typedef __attribute__((address_space(1))) v4i* gv4i_p;   // "__device__"
    typedef __attribute__((address_space(3))) v4i* lv4i_p;   // "__shared__"
    #pragma unroll
    for (int i = 0; i < kChunks; ++i) {
      const int idx = i * kThreads + tid;
      __builtin_amdgcn_global_load_async_to_lds_b128(
          (gv4i_p)(g + idx), (lv4i_p)(sdata + idx), 0, 0);
    }
  }
#endif

  // ---------------- Pass 1: rowsum of sigmoid + per-lane max -------------
  float s0 = 0.f, s1 = 0.f, s2 = 0.f, s3 = 0.f;
  float m0 = -__builtin_inff(), m1 = -__builtin_inff();
  float m2 = -__builtin_inff(), m3 = -__builtin_inff();

  auto accum = [&](float4 v) {
    s0 += sig_fast(v.x); s1 += sig_fast(v.y);
    s2 += sig_fast(v.z); s3 += sig_fast(v.w);
    m0 = fmaxf(m0, v.x); m1 = fmaxf(m1, v.y);
    m2 = fmaxf(m2, v.z); m3 = fmaxf(m3, v.w);
  };

#if HAS_ASYNC_LDS
  // Async loads complete in order per wave: process in groups of 8 chunks.
  WAIT_ASYNC(24);
  #pragma unroll
  for (int i = 0;  i < 8;  ++i) accum(sdata[i * kThreads + tid]);
  WAIT_ASYNC(16);
  #pragma unroll
  for (int i = 8;  i < 16; ++i) accum(sdata[i * kThreads + tid]);
  WAIT_ASYNC(8);
  #pragma unroll
  for (int i = 16; i < 24; ++i) accum(sdata[i * kThreads + tid]);
  WAIT_ASYNC(0);
  #pragma unroll
  for (int i = 24; i < 32; ++i) accum(sdata[i * kThreads + tid]);
#else
  for (int i = 0; i < kChunks; ++i) accum(g[i * kThreads + tid]);
#endif

  float sum = (s0 + s1) + (s2 + s3);
  float mx  = fmaxf(fmaxf(m0, m1), fmaxf(m2, m3));

  {
    float ws = wave_sum(sum);
    if (lane == 0) s_sum[wave] = ws;
  }

  // Wave threshold: 10th largest of this wave's 32 lane-maxima.
  // Each lane-max is a real row element => threshold <= true row 10th value.
  float th = -__builtin_inff();
  {
    float cand = mx;
    #pragma unroll
    for (int k = 0; k < kK; ++k) {
      float M = wave_max(cand);
      unsigned long long b = __ballot(cand == M);
      if (lane == __builtin_ctzll(b)) cand = -__builtin_inff();
      th = M;
    }
  }

  // ---------------- Pass 2: exact top-10 above threshold (from LDS) ------
  float t10[kK];
  #pragma unroll
  for (int k = 0; k < kK; ++k) t10[k] = -__builtin_inff();

  auto consider = [&](float x) {
    if (x >= th) {                       // rare -> bubble almost never runs
      #pragma unroll
      for (int k = 0; k < kK; ++k) {     // sorted insert, registers only
        float hi = fmaxf(t10[k], x);
        x        = fminf(t10[k], x);
        t10[k]   = hi;
      }
    }
  };

  for (int i = 0; i < kChunks; ++i) {
#if HAS_ASYNC_LDS
    float4 v = sdata[i * kThreads + tid];     // zero extra HBM traffic
#else
    float4 v = g[i * kThreads + tid];         // re-read (should hit L2)
#endif
    float cm = fmaxf(fmaxf(v.x, v.y), fmaxf(v.z, v.w));
    if (cm >= th) { consider(v.x); consider(v.y); consider(v.z); consider(v.w); }
  }

  // Wave-level merge: 10 rounds of shuffle argmax; lane k keeps k-th value.
  {
    float mine = -__builtin_inff();
    #pragma unroll
    for (int k = 0; k < kK; ++k) {
      float cand = t10[0];
      float M = wave_max(cand);
      unsigned long long b = __ballot(cand == M);
      if (lane == __builtin_ctzll(b)) {
        t10[0] = t10[1]; t10[1] = t10[2]; t10[2] = t10[3]; t10[3] = t10[4];
        t10[4] = t10[5]; t10[5] = t10[6]; t10[6] = t10[7]; t10[7] = t10[8];
        t10[8] = t10[9]; t10[9] = -__builtin_inff();
      }
      if (lane == k) mine = M;
    }
    if (lane < kK) s_top[wave * kK + lane] = mine;
  }

  __syncthreads();

  // Block-level merge of 8 sorted wave lists + final per-row term.
  if (tid == 0) {
    float rowsum = 0.0f;
    #pragma unroll
    for (int w = 0; w < kWaves; ++w) rowsum += s_sum[w];

    int p[kWaves] = {0, 0, 0, 0, 0, 0, 0, 0};
    float tops = 0.0f;
    for (int k = 0; k < kK; ++k) {
      float bv = -__builtin_inff(); int bw = 0;
      for (int w = 0; w < kWaves; ++w) {
        if (p[w] < kK) {
          float c = s_top[w * kK + p[w]];
          if (c > bv) { bv = c; bw = w; }
        }
      }
      p[bw]++;
      tops += sig_fast(bv);
    }
    float r = tops / rowsum;
    terms[row] = r * logf(r + 1e-10f);
  }
}

// Deterministic final reduction (fixed-order tree; no float atomics).
__global__ void __launch_bounds__(256)
finalize_kernel(const float* __restrict__ terms, float* __restrict__ out) {
  __shared__ float red[256];
  const int tid = threadIdx.x;
  float s = 0.0f;
  for (int i = tid; i < kB; i += 256) s += terms[i];
  red[tid] = s;
  __syncthreads();
  for (int off = 128; off > 0; off >>= 1) {
    if (tid < off) red[tid] += red[tid + off];
    __syncthreads();
  }
  if (tid == 0) out[0] = -red[0] / (float)kB;
}

extern "C" void kernel_launch(void* const* d_in, const int* in_sizes, int n_in,
                              void* d_out, int out_size, void* d_ws, size_t ws_size,
                              hipStream_t stream) {
  (void)in_sizes; (void)n_in; (void)out_size; (void)ws_size;
  const float* S   = (const float*)d_in[0];
  float* out       = (float*)d_out;
  float* terms     = (float*)d_ws;            // 2048 floats of scratch

  // Row buffer is 128 KB of dynamic LDS (CDNA5 WGP has 320 KB).
  (void)hipFuncSetAttribute(reinterpret_cast<const void*>(row_loss_kernel),
                            hipFuncAttributeMaxDynamicSharedMemorySize,
                            kN * (int)sizeof(float));

  row_loss_kernel<<<kB, kThreads, kN * sizeof(float), stream>>>(S, terms);
  finalize_kernel<<<1, 256, 0, stream>>>(terms, out);
}